// SNNRegLossExact_21457656611490
// MI455X (gfx1250) — compile-verified
//
#include <hip/hip_runtime.h>
#include <hip/hip_bf16.h>
#include <stdint.h>

typedef __attribute__((ext_vector_type(2))) float v2f;
typedef __attribute__((ext_vector_type(8))) float v8f;

#define THRESH    0.05f
#define LAM2      0.5f
#define EPS_F     1e-08f
#define T_DEFAULT 2.0f

// ---------------------------------------------------------------------------
// 1) Per-column mean / std (ddof=1), one block per column.
// ---------------------------------------------------------------------------
__global__ void colstats_kernel(const float* __restrict__ x, float* __restrict__ m,
                                float* __restrict__ s, int Bn, int Dn) {
    __shared__ float ssum[256], ssq[256];
    int c = blockIdx.x;
    float sum = 0.f, sq = 0.f;
    for (int i = threadIdx.x; i < Bn; i += blockDim.x) {
        float v = x[(size_t)i * Dn + c];
        sum += v; sq += v * v;
    }
    ssum[threadIdx.x] = sum; ssq[threadIdx.x] = sq;
    __syncthreads();
    for (int st = 128; st >= 1; st >>= 1) {
        if (threadIdx.x < st) { ssum[threadIdx.x] += ssum[threadIdx.x + st];
                                ssq[threadIdx.x]  += ssq[threadIdx.x + st]; }
        __syncthreads();
    }
    if (threadIdx.x == 0) {
        float mean = ssum[0] / (float)Bn;
        float var  = (ssq[0] - (float)Bn * mean * mean) / (float)(Bn - 1);
        float sd   = sqrtf(fmaxf(var, 0.f));
        m[c] = mean;
        s[c] = fmaxf(sd, 1e-06f);
    }
}

// ---------------------------------------------------------------------------
// 2) Normalize rows, extract z2 (col 1), zero col 1 in Xn so the K=128 GEMM
//    equals the 127-dim sum. Also produce row_sq and zero all accumulators.
//    One block (128 threads) per row.
// ---------------------------------------------------------------------------
__global__ void rownorm_kernel(const float* __restrict__ x, const float* __restrict__ m,
                               const float* __restrict__ s, float* __restrict__ Xn,
                               float* __restrict__ z2, float* __restrict__ rsq,
                               float* __restrict__ num, float* __restrict__ den1,
                               float* __restrict__ den2, uint32_t* __restrict__ state,
                               int Dn) {
    __shared__ float sh[128];
    int row = blockIdx.x, c = threadIdx.x;
    float v = (x[(size_t)row * Dn + c] - m[c]) / s[c];
    if (c == 1) { z2[row] = v; v = 0.f; }          // TARGET_DIM = 1
    Xn[(size_t)row * Dn + c] = v;
    sh[c] = v * v;
    __syncthreads();
    for (int st = 64; st >= 1; st >>= 1) {
        if (c < st) sh[c] += sh[c + st];
        __syncthreads();
    }
    if (c == 0) {
        rsq[row] = sh[0];
        num[row] = 0.f; den1[row] = 0.f; den2[row] = 0.f;
    }
    if (row == 0 && c < 4) state[c] = 0u;          // npos, prefix, krem, T
}

// ---------------------------------------------------------------------------
// 3) Radix-select median of d2 over same_age pairs (3 passes over float bits).
//    state[0]=n_pos, state[1]=prefix, state[2]=krem, state[3]=T_eff bits.
// ---------------------------------------------------------------------------
__global__ void zero_kernel(uint32_t* __restrict__ p, int n) {
    int i = blockIdx.x * blockDim.x + threadIdx.x;
    if (i < n) p[i] = 0u;
}

__global__ void count_kernel(const float* __restrict__ z2, const float* __restrict__ y,
                             uint32_t* __restrict__ hist, uint32_t* __restrict__ npr,
                             uint32_t* __restrict__ state, int level, int nbuckets, int Bn) {
    __shared__ uint32_t sh[2048];
    __shared__ uint32_t srow;
    int i = blockIdx.x;
    for (int t = threadIdx.x; t < nbuckets; t += blockDim.x) sh[t] = 0u;
    if (threadIdx.x == 0) srow = 0u;
    __syncthreads();
    float yi = y[i], zi = z2[i];
    uint32_t prefix = state[1];
    uint32_t cnt = 0;
    for (int j = threadIdx.x; j < Bn; j += blockDim.x) {
        if (j == i) continue;
        if (fabsf(yi - y[j]) <= THRESH) {
            float dz = zi - z2[j];
            uint32_t bits = __float_as_uint(dz * dz);   // >= 0: uint order == float order
            cnt++;
            bool ok; uint32_t b;
            if (level == 0)      { ok = true;                        b = bits >> 21; }
            else if (level == 1) { ok = ((bits >> 21) == prefix);    b = (bits >> 10) & 0x7FFu; }
            else                 { ok = ((bits >> 10) == prefix);    b = bits & 0x3FFu; }
            if (ok) atomicAdd(&sh[b], 1u);
        }
    }
    if (level == 0 && cnt) atomicAdd(&srow, cnt);
    __syncthreads();
    if (level == 0 && threadIdx.x == 0) {
        npr[i] = srow;                       // per-row positive count (for has_pos)
        if (srow) atomicAdd(&state[0], srow);
    }
    for (int t = threadIdx.x; t < nbuckets; t += blockDim.x)
        if (sh[t]) atomicAdd(&hist[t], sh[t]);
}

__global__ void scan_kernel(const uint32_t* __restrict__ hist, uint32_t* __restrict__ state,
                            int level, int nbuckets) {
    uint32_t npos = state[0];
    uint32_t krem = (level == 0) ? ((npos > 0 ? npos - 1u : 0u) >> 1) : state[2];
    uint32_t cum = 0, sel = 0;
    for (int b = 0; b < nbuckets; ++b) {
        uint32_t h = hist[b];
        if (cum + h > krem) { sel = (uint32_t)b; break; }
        cum += h;
    }
    krem -= cum;
    uint32_t prefix = state[1];
    if (level == 0)       prefix = sel;
    else if (level == 1)  prefix = (prefix << 11) | sel;
    else {
        uint32_t med_bits = (prefix << 10) | sel;
        float med = __uint_as_float(med_bits);
        float T = (npos > 0) ? fmaxf(med, 1e-06f) : T_DEFAULT;
        state[3] = __float_as_uint(T);
    }
    state[1] = prefix;
    state[2] = krem;
}

// ---------------------------------------------------------------------------
// 4) Gram matrix via V_WMMA_F32_16X16X4_F32 with fused kernel epilogue.
//    One wave per 16x16 output tile; K loop of 32 wmma ops (K=128).
//    A layout: lane = {M=lane%16, K pair 2g,2g+1}; B mirrored on the j block.
//    D layout: acc[r] is element (M = r + 8*(lane/16), N = lane%16).
// ---------------------------------------------------------------------------
__global__ __launch_bounds__(256)
void gram_kernel(const float* __restrict__ Xn, const float* __restrict__ z2,
                 const float* __restrict__ y, const float* __restrict__ rsq,
                 const uint32_t* __restrict__ state, float* __restrict__ num,
                 float* __restrict__ den1, float* __restrict__ den2, int Bn, int Dn) {
    const float T    = __uint_as_float(state[3]);
    const float invT = 1.0f / T;
    const float invM = 1.0f / (float)(Dn - 1);      // mean over the 127 kept dims

    int wid  = (blockIdx.x * blockDim.x + threadIdx.x) >> 5;
    int lane = threadIdx.x & 31;
    int ntj  = Bn >> 4;
    int tiles = ntj * ntj;
    if (wid >= tiles) return;                        // uniform per-wave: EXEC stays all-1
    int ti = wid / ntj, tj = wid % ntj;
    int g  = lane >> 4;                              // half-wave (K / M-half select)
    int lm = lane & 15;
    int i0 = ti << 4, j0 = tj << 4;

    const float* Ap = Xn + (size_t)(i0 + lm) * Dn + 2 * g;
    const float* Bp = Xn + (size_t)(j0 + lm) * Dn + 2 * g;

    v8f acc = {};
    #pragma unroll 8
    for (int kk = 0; kk < Dn; kk += 4) {
        v2f a = *(const v2f*)(Ap + kk);
        v2f b = *(const v2f*)(Bp + kk);
        acc = __builtin_amdgcn_wmma_f32_16x16x4_f32(
            /*neg_a=*/false, a, /*neg_b=*/false, b,
            /*c_mod=*/(short)0, acc, /*reuse_a=*/false, /*reuse_b=*/false);
    }

    int   j  = j0 + lm;
    float yj = y[j], zj = z2[j], rj = rsq[j];
    #pragma unroll
    for (int r = 0; r < 8; ++r) {
        int   i  = i0 + r + 8 * g;
        float yi = y[i], zi = z2[i], ri = rsq[i];
        bool  offd = (i != j);
        bool  same = offd && (fabsf(yi - yj) <= THRESH);
        float dz  = zi - zj;
        float K2  = expf(-(dz * dz) * invT);
        float sqm = (ri + rj - 2.0f * acc[r]) * invM;
        float Ko  = expf(-sqm * invT);
        float nc  = same ? K2 : 0.f;
        float d1  = offd ? K2 : 0.f;
        float d2c = same ? Ko : 0.f;
        // reduce over the 16 columns held by this half-wave (same row i)
        #pragma unroll
        for (int msk = 8; msk >= 1; msk >>= 1) {
            nc  += __shfl_xor(nc,  msk, 16);
            d1  += __shfl_xor(d1,  msk, 16);
            d2c += __shfl_xor(d2c, msk, 16);
        }
        if (lm == 0) {
            atomicAdd(&num[i],  nc);
            atomicAdd(&den1[i], d1);
            atomicAdd(&den2[i], d2c);
        }
    }
}

// ---------------------------------------------------------------------------
// 5) Final per-row loss + reduction to the scalar output.
// ---------------------------------------------------------------------------
__global__ void finalize_kernel(const float* __restrict__ num, const float* __restrict__ den1,
                                const float* __restrict__ den2, const uint32_t* __restrict__ npr,
                                float* __restrict__ out, int Bn) {
    __shared__ float    sl[256];
    __shared__ uint32_t sc[256];
    float loss = 0.f; uint32_t cnt = 0;
    for (int i = threadIdx.x; i < Bn; i += blockDim.x) {
        if (npr[i] > 0) {
            float denom = den1[i] + LAM2 * den2[i] + EPS_F;
            float frac  = num[i] / denom;
            frac = fminf(fmaxf(frac, 1e-12f), 1.0f - 1e-07f);
            loss += -logf(frac);
            cnt++;
        }
    }
    sl[threadIdx.x] = loss; sc[threadIdx.x] = cnt;
    __syncthreads();
    for (int st = 128; st >= 1; st >>= 1) {
        if (threadIdx.x < st) { sl[threadIdx.x] += sl[threadIdx.x + st];
                                sc[threadIdx.x] += sc[threadIdx.x + st]; }
        __syncthreads();
    }
    if (threadIdx.x == 0)
        out[0] = (sc[0] > 0) ? sl[0] / (float)sc[0] : 0.f;
}

// ---------------------------------------------------------------------------
extern "C" void kernel_launch(void* const* d_in, const int* in_sizes, int n_in,
                              void* d_out, int out_size, void* d_ws, size_t ws_size,
                              hipStream_t stream) {
    const float* x = (const float*)d_in[0];
    const float* y = (const float*)d_in[1];
    float* out = (float*)d_out;

    const int Bn = in_sizes[1];          // 1024
    const int Dn = in_sizes[0] / Bn;     // 128

    // workspace carve-out (floats / uints)
    float* ws   = (float*)d_ws;
    float* m    = ws;
    float* s    = m + Dn;
    float* Xn   = s + Dn;
    float* z2   = Xn + (size_t)Bn * Dn;
    float* rsq  = z2 + Bn;
    float* num  = rsq + Bn;
    float* den1 = num + Bn;
    float* den2 = den1 + Bn;
    uint32_t* npr   = (uint32_t*)(den2 + Bn);
    uint32_t* hist  = npr + Bn;
    uint32_t* state = hist + 2048;

    colstats_kernel<<<Dn, 256, 0, stream>>>(x, m, s, Bn, Dn);
    rownorm_kernel<<<Bn, Dn, 0, stream>>>(x, m, s, Xn, z2, rsq, num, den1, den2, state, Dn);

    // exact median via 3-level radix select on float bits of d2
    zero_kernel<<<8, 256, 0, stream>>>(hist, 2048);
    count_kernel<<<Bn, 256, 0, stream>>>(z2, y, hist, npr, state, 0, 2048, Bn);
    scan_kernel<<<1, 1, 0, stream>>>(hist, state, 0, 2048);
    zero_kernel<<<8, 256, 0, stream>>>(hist, 2048);
    count_kernel<<<Bn, 256, 0, stream>>>(z2, y, hist, npr, state, 1, 2048, Bn);
    scan_kernel<<<1, 1, 0, stream>>>(hist, state, 1, 2048);
    zero_kernel<<<4, 256, 0, stream>>>(hist, 1024);
    count_kernel<<<Bn, 256, 0, stream>>>(z2, y, hist, npr, state, 2, 1024, Bn);
    scan_kernel<<<1, 1, 0, stream>>>(hist, state, 2, 1024);

    // WMMA Gram matrix + fused mask/exp/reduce epilogue
    int ntj   = Bn / 16;
    int tiles = ntj * ntj;               // 4096 tiles, 1 wave each
    int blocks = (tiles + 7) / 8;        // 8 waves (256 threads) per block
    gram_kernel<<<blocks, 256, 0, stream>>>(Xn, z2, y, rsq, state, num, den1, den2, Bn, Dn);

    finalize_kernel<<<1, 256, 0, stream>>>(num, den1, den2, npr, out, Bn);
}